// CRF_solo_51204600103385
// MI455X (gfx1250) — compile-verified
//
#include <hip/hip_runtime.h>

typedef __attribute__((ext_vector_type(2))) float v2f;
typedef __attribute__((ext_vector_type(8))) float v8f;

#define V_ 50257
#define E_ 256
#define C_ 48
#define B_ 64
#define T_ 1024
#define LDSB 258   // padded LDS row stride (floats) -> conflict-free float2 reads

// ---------------------------------------------------------------------------
// Kernel 1: fused embedding gather + logits GEMM via V_WMMA_F32_16X16X4_F32.
// One wave = one 16-row M-tile x all 3 N-tiles (N=48). K=256 in steps of 4.
// Memory-bound: ~77MB traffic vs 1.6 GFLOP, so exact f32 WMMA is optimal.
// ---------------------------------------------------------------------------
__global__ __launch_bounds__(128) void logits_wmma_kernel(
    const int* __restrict__ x, const float* __restrict__ emb,
    const float* __restrict__ fcw, const float* __restrict__ fcb,
    float* __restrict__ logits)
{
    __shared__ float bs[C_ * LDSB];   // fc_w staged, padded stride
    const int tid = threadIdx.x;

    // Cooperative coalesced stage of fc_w (48x256 f32) into LDS.
    for (int idx = tid; idx < C_ * (E_ / 4); idx += 128) {
        const int row = idx >> 6;            // idx / 64
        const int seg = (idx & 63) * 4;
        const float4 v = *(const float4*)(fcw + row * E_ + seg);
        float* p = &bs[row * LDSB + seg];
        p[0] = v.x; p[1] = v.y; p[2] = v.z; p[3] = v.w;
    }
    __syncthreads();

    const int wave = tid >> 5;
    const int lane = tid & 31;
    const int half = lane >> 4;              // K-pair selector (ISA A/B layout)
    const int l16  = lane & 15;
    const int tile_m = blockIdx.x * 4 + wave;
    const int m0 = tile_m * 16;

    // A: gathered embedding row in WMMA A layout (lane -> M=l16, K-pair=2*half)
    const int arow = x[m0 + l16];
    const float* aptr  = emb + (size_t)arow * E_ + 2 * half;
    // B: fc_w rows from LDS in WMMA B layout (lane -> N=l16, K-pair=2*half)
    const float* bptr0 = &bs[(0 * 16 + l16) * LDSB + 2 * half];
    const float* bptr1 = &bs[(1 * 16 + l16) * LDSB + 2 * half];
    const float* bptr2 = &bs[(2 * 16 + l16) * LDSB + 2 * half];

    v8f c0 = {0.f, 0.f, 0.f, 0.f, 0.f, 0.f, 0.f, 0.f};
    v8f c1 = c0, c2 = c0;

    for (int k0 = 0; k0 < E_; k0 += 4) {
        const v2f a  = *(const v2f*)(aptr  + k0);
        const v2f b0 = *(const v2f*)(bptr0 + k0);
        const v2f b1 = *(const v2f*)(bptr1 + k0);
        const v2f b2 = *(const v2f*)(bptr2 + k0);
        c0 = __builtin_amdgcn_wmma_f32_16x16x4_f32(false, a, false, b0, (short)0, c0, false, false);
        c1 = __builtin_amdgcn_wmma_f32_16x16x4_f32(false, a, false, b1, (short)0, c1, false, false);
        c2 = __builtin_amdgcn_wmma_f32_16x16x4_f32(false, a, false, b2, (short)0, c2, false, false);
    }

    // Epilogue: bias add + store. C/D layout: row = r + 8*half, col = l16.
    const float bias0 = fcb[ 0 + l16];
    const float bias1 = fcb[16 + l16];
    const float bias2 = fcb[32 + l16];
    #pragma unroll
    for (int r = 0; r < 8; ++r) {
        const int m = m0 + r + 8 * half;
        float* o = logits + (size_t)m * C_;
        o[ 0 + l16] = c0[r] + bias0;
        o[16 + l16] = c1[r] + bias1;
        o[32 + l16] = c2[r] + bias2;
    }
}

// ---------------------------------------------------------------------------
// Kernel 2: CRF forward scan + numerator + loss. One block per batch element,
// lanes 0..47 own destination class j; trans column hoisted into 48 VGPRs.
// ---------------------------------------------------------------------------
__global__ __launch_bounds__(64) void crf_scan_kernel(
    const float* __restrict__ logits, const int* __restrict__ labels,
    const float* __restrict__ start_trans, const float* __restrict__ end_trans,
    const float* __restrict__ trans, float* __restrict__ out_scalar)
{
    __shared__ float alpha_s[C_];
    __shared__ float red[64];

    const int b   = blockIdx.x;
    const int tid = threadIdx.x;
    const int j   = tid;
    const bool act = (j < C_);

    const float* lg = logits + (size_t)b * T_ * C_;
    const int*   lb = labels + (size_t)b * T_;

    // Hoist trans[:, j] into registers (fully unrolled -> stays in VGPRs).
    float trc[C_];
    if (act) {
        #pragma unroll
        for (int i = 0; i < C_; ++i) trc[i] = trans[i * C_ + j];
        alpha_s[j] = start_trans[j] + lg[j];   // alpha0
    }

    // Numerator partial sums (emit + label transitions), t-strided over 64 threads.
    float part = 0.f;
    for (int t = tid; t < T_; t += 64) {
        part += lg[t * C_ + lb[t]];
        if (t < T_ - 1) part += trans[lb[t] * C_ + lb[t + 1]];
    }
    red[tid] = part;
    __syncthreads();

    // Sequential forward recursion over T.
    for (int t = 1; t < T_; ++t) {
        float na = 0.f;
        if (act) {
            float vals[C_];
            float m = -3.4e38f;
            #pragma unroll
            for (int i = 0; i < C_; ++i) {
                const float v = alpha_s[i] + trc[i];
                vals[i] = v;
                m = fmaxf(m, v);
            }
            float s = 0.f;
            #pragma unroll
            for (int i = 0; i < C_; ++i) s += __expf(vals[i] - m);
            na = m + __logf(s) + lg[t * C_ + j];
        }
        __syncthreads();
        if (act) alpha_s[j] = na;
        __syncthreads();
    }

    if (tid == 0) {
        float numer = 0.f;
        for (int i = 0; i < 64; ++i) numer += red[i];
        numer += start_trans[lb[0]] + end_trans[lb[T_ - 1]];

        float m = -3.4e38f;
        for (int i = 0; i < C_; ++i) m = fmaxf(m, alpha_s[i] + end_trans[i]);
        float s = 0.f;
        for (int i = 0; i < C_; ++i) s += __expf(alpha_s[i] + end_trans[i] - m);
        const float log_z = m + __logf(s);

        // loss = -llh = sum_b (log_z_b - numerator_b)
        atomicAdd(out_scalar, log_z - numer);
    }
}

__global__ void zero_scalar_kernel(float* __restrict__ p) { *p = 0.f; }

extern "C" void kernel_launch(void* const* d_in, const int* in_sizes, int n_in,
                              void* d_out, int out_size, void* d_ws, size_t ws_size,
                              hipStream_t stream) {
    const int*   x      = (const int*)  d_in[0];
    const int*   labels = (const int*)  d_in[1];
    const float* emb    = (const float*)d_in[2];
    const float* fcw    = (const float*)d_in[3];
    const float* fcb    = (const float*)d_in[4];
    const float* st     = (const float*)d_in[5];
    const float* et     = (const float*)d_in[6];
    const float* tr     = (const float*)d_in[7];

    float* logits = (float*)d_out;
    float* loss   = logits + (size_t)B_ * T_ * C_;   // scalar -llh slot

    // 65536 rows / 16 per wave / 4 waves per block = 1024 blocks.
    logits_wmma_kernel<<<dim3(1024), dim3(128), 0, stream>>>(x, emb, fcw, fcb, logits);
    zero_scalar_kernel<<<dim3(1), dim3(1), 0, stream>>>(loss);
    crf_scan_kernel<<<dim3(B_), dim3(64), 0, stream>>>(logits, labels, st, et, tr, loss);
}